// StepwiseMLPHead_44281112822377
// MI455X (gfx1250) — compile-verified
//
#include <hip/hip_runtime.h>

// ---- problem constants (match reference) ----
constexpr int kInputDim = 1024;
constexpr int kDayEmb   = 8;
constexpr int kPredLen  = 96;
constexpr int kHidden   = 512;
constexpr int kBatch    = 4096;

// ---- tiling ----
constexpr int kBTile  = 64;   // batch rows per block
constexpr int kKC     = 16;   // K chunk staged in LDS
constexpr int kNWaves = 8;    // 256 threads / wave32
constexpr int kWaveN  = kHidden / kNWaves;  // 64 columns per wave (4 N-tiles)

typedef __attribute__((ext_vector_type(2))) float v2f;
typedef __attribute__((ext_vector_type(8))) float v8f;

__global__ __launch_bounds__(256) void smlp_fused_kernel(
    const float* __restrict__ x,        // [4096][1024]
    const float* __restrict__ day_emb,  // [96][8]
    const float* __restrict__ W1,       // [96][1032][512]
    const float* __restrict__ b1,       // [96][512]
    const float* __restrict__ W2,       // [96][512]
    const float* __restrict__ b2,       // [96]
    float* __restrict__ out)            // [4096][96]
{
  __shared__ float sA[kBTile * kKC];        // x tile [64][16]                 4 KB
  // W1 tile, K-pair interleaved: sB[pair][n][2] with pair = k/2 (8 pairs)
  // so {B[k][n], B[k+1][n]} is one aligned b64 -> single ds_load_b64 fragment
  __shared__ float sB[(kKC / 2) * kHidden * 2];  //                           32 KB
  __shared__ float sBias[kHidden];          //                                 2 KB
  __shared__ float sRed[kBTile * kNWaves];  //                                 2 KB

  const int p    = blockIdx.y;
  const int b0   = blockIdx.x * kBTile;
  const int tid  = threadIdx.x;
  const int lane = tid & 31;
  const int wave = tid >> 5;
  const int l16  = lane & 15;
  const int hl   = lane >> 4;   // lane half: 0 = lanes 0-15, 1 = lanes 16-31

  const float* __restrict__ W1p = W1 + (size_t)p * (kInputDim + kDayEmb) * kHidden;

  // --- per-p bias vector: day_emb[p] . W1e[p] + b1[p]  (512 values into LDS) ---
  for (int h = tid; h < kHidden; h += 256) {
    float s = b1[p * kHidden + h];
#pragma unroll
    for (int e = 0; e < kDayEmb; ++e)
      s += day_emb[p * kDayEmb + e] * W1p[(size_t)(kInputDim + e) * kHidden + h];
    sBias[h] = s;
  }

  // 4 M-tiles x 4 N-tiles of 16x16 f32 accumulators per wave
  v8f acc[4][4] = {};

  for (int k0 = 0; k0 < kInputDim; k0 += kKC) {
    __syncthreads();

    // stage x tile: 64x16 floats = 256 float4, one per thread (row-major)
    {
      const int row = tid >> 2;          // 0..63
      const int q   = (tid & 3) << 2;    // 0,4,8,12
      const float4 v = *(const float4*)&x[(size_t)(b0 + row) * kInputDim + k0 + q];
      *(float4*)&sA[row * kKC + q] = v;
    }
    // stage W1 tile K-pair interleaved: 8 pairs x 512 cols, 4 units per thread
    // unit = (row pair, 4 consecutive n): 2 global float4 -> 2 interleaved float4
#pragma unroll
    for (int i = 0; i < 4; ++i) {
      const int u    = tid + i * 256;       // 0..1023
      const int pair = u >> 7;              // 0..7
      const int n    = (u & 127) << 2;      // 0..508, step 4
      const float4 r0 = *(const float4*)&W1p[(size_t)(k0 + 2 * pair + 0) * kHidden + n];
      const float4 r1 = *(const float4*)&W1p[(size_t)(k0 + 2 * pair + 1) * kHidden + n];
      const float4 w0 = make_float4(r0.x, r1.x, r0.y, r1.y);
      const float4 w1 = make_float4(r0.z, r1.z, r0.w, r1.w);
      *(float4*)&sB[pair * (kHidden * 2) + 2 * n + 0] = w0;
      *(float4*)&sB[pair * (kHidden * 2) + 2 * n + 4] = w1;
    }
    // prefetch next W1 chunk into cache (global_prefetch_b8), 128B/lane strides
    if (k0 + kKC < kInputDim)
      __builtin_prefetch(&W1p[(size_t)(k0 + kKC) * kHidden + tid * 32], 0, 1);

    __syncthreads();

#pragma unroll
    for (int kk = 0; kk < kKC; kk += 4) {
      // A fragments (16x4 f32): lane l16 = row M, K pair {0,1} or {2,3} by half
      v2f afrag[4];
#pragma unroll
      for (int mt = 0; mt < 4; ++mt)
        afrag[mt] = *(const v2f*)&sA[(mt * 16 + l16) * kKC + kk + 2 * hl];

      const int kp = (kk >> 1) + hl;        // K-pair index into sB
#pragma unroll
      for (int nt = 0; nt < 4; ++nt) {
        const int n = wave * kWaveN + nt * 16 + l16;
        // single aligned b64: {B[2*kp][n], B[2*kp+1][n]}
        const v2f bfrag = *(const v2f*)&sB[kp * (kHidden * 2) + 2 * n];
#pragma unroll
        for (int mt = 0; mt < 4; ++mt)
          acc[mt][nt] = __builtin_amdgcn_wmma_f32_16x16x4_f32(
              false, afrag[mt], false, bfrag, (short)0, acc[mt][nt],
              false, false);
      }
    }
  }

  __syncthreads();

  // --- fused epilogue: relu(acc + bias) . W2, reduced over HIDDEN ---
  const float* __restrict__ W2p = W2 + (size_t)p * kHidden;
#pragma unroll
  for (int mt = 0; mt < 4; ++mt) {
    float rs[8];
#pragma unroll
    for (int r = 0; r < 8; ++r) rs[r] = 0.0f;

#pragma unroll
    for (int nt = 0; nt < 4; ++nt) {
      const int n     = wave * kWaveN + nt * 16 + l16;
      const float bn  = sBias[n];
      const float w2n = W2p[n];
#pragma unroll
      for (int r = 0; r < 8; ++r) {
        // C/D layout: VGPR r, lanes 0-15 -> row mt*16+r, lanes 16-31 -> +8
        float v = acc[mt][nt][r] + bn;
        v = fmaxf(v, 0.0f);
        rs[r] += v * w2n;
      }
    }
    // reduce across the 16 lanes of each half (each half = one row set)
#pragma unroll
    for (int r = 0; r < 8; ++r) {
      float s = rs[r];
      s += __shfl_xor(s, 1, 16);
      s += __shfl_xor(s, 2, 16);
      s += __shfl_xor(s, 4, 16);
      s += __shfl_xor(s, 8, 16);
      if (l16 == 0)
        sRed[(mt * 16 + hl * 8 + r) * kNWaves + wave] = s;
    }
  }

  __syncthreads();

  // cross-wave reduction + b2, strided store out[b][p]
  if (tid < kBTile) {
    float s = b2[p];
#pragma unroll
    for (int w = 0; w < kNWaves; ++w) s += sRed[tid * kNWaves + w];
    out[(size_t)(b0 + tid) * kPredLen + p] = s;
  }
}

extern "C" void kernel_launch(void* const* d_in, const int* in_sizes, int n_in,
                              void* d_out, int out_size, void* d_ws, size_t ws_size,
                              hipStream_t stream) {
  const float* x       = (const float*)d_in[0];
  const float* day_emb = (const float*)d_in[1];
  const float* W1      = (const float*)d_in[2];
  const float* b1      = (const float*)d_in[3];
  const float* W2      = (const float*)d_in[4];
  const float* b2      = (const float*)d_in[5];
  float* out           = (float*)d_out;

  dim3 grid(kBatch / kBTile, kPredLen);  // 64 x 96 blocks
  smlp_fused_kernel<<<grid, 256, 0, stream>>>(x, day_emb, W1, b1, W2, b2, out);
}